// Motion_Relation_Mining_73950746903016
// MI455X (gfx1250) — compile-verified
//
#include <hip/hip_runtime.h>
#include <math.h>

// Motion_Relation_Mining for MI455X (gfx1250, wave32).
//
//   mag_kernel   : bandwidth-critical streaming pass (b128 loads), channel-sum
//                  of frame diffs into d_ws. The per-batch MI "corr" term is a
//                  uniform additive shift per (b,f) slice and cannot change
//                  top_k ordering -> skipped (halves memory traffic).
//   slice_kernel : one 256-thread workgroup per (b,f) slice. Stages the whole
//                  256KB mag slice into LDS (320KB/WGP on CDNA5):
//                    rows   0..127 : TENSOR_LOAD_TO_LDS (TDM, D# descriptor)
//                    rows 128..255 : global_load_async_to_lds_b128 (ASYNCcnt)
//                  then: radix-select exact 64th-largest -> bitonic top-64
//                  (stable ties) -> centers -> 2x2 cov inverse -> maha via
//                  V_WMMA_F32_16X16X4_F32 -> radix-select 169 smallest -> graph.

#define B_      4
#define T_      8
#define CCH     16
#define HW      65536
#define NF      7
#define NSLICE  28
#define PP      65536
#define RTOP    64
#define TAUSQ   169

typedef __attribute__((ext_vector_type(2))) float v2f;
typedef __attribute__((ext_vector_type(8))) float v8f;
typedef __attribute__((ext_vector_type(4))) unsigned int u32x4;
typedef __attribute__((ext_vector_type(4))) int i32x4v;
typedef __attribute__((ext_vector_type(8))) int i32x8v;

__device__ __forceinline__ unsigned keyAsc(float x) {
  unsigned b = __float_as_uint(x);
  return (b & 0x80000000u) ? ~b : (b | 0x80000000u);
}

// ---------------------------------------------------------------------------
// Kernel 1: mag[slice][p] = sum_c (x[b,f+1,c,h,w] - x[b,f,c,h,w])
// ---------------------------------------------------------------------------
__global__ void __launch_bounds__(256)
mag_kernel(const float* __restrict__ x, float* __restrict__ mag) {
  const int gid = blockIdx.x * 256 + threadIdx.x;   // 28 * 16384 threads
  const int slice = gid >> 14;
  if (slice >= NSLICE) return;
  const int e4  = gid & 16383;
  const int b   = slice / NF;
  const int f   = slice - b * NF;
  const int pix = e4 << 2;
  const int h   = pix >> 8;
  const int w   = pix & 255;

  const float* base0 = x + ((size_t)(b * T_ + f) * CCH) * HW + (size_t)h * 256 + w;
  const float* base1 = base0 + (size_t)CCH * HW;

  float sx = 0.f, sy = 0.f, sz = 0.f, sw = 0.f;
#pragma unroll
  for (int c = 0; c < CCH; ++c) {
    const float4 a1 = *(const float4*)(base1 + (size_t)c * HW);
    const float4 a0 = *(const float4*)(base0 + (size_t)c * HW);
    sx += a1.x - a0.x;  sy += a1.y - a0.y;
    sz += a1.z - a0.z;  sw += a1.w - a0.w;
  }
  const int p = (((h >> 3) * 32 + (w >> 3)) << 6) + ((h & 7) << 3) + (w & 7);
  float4 o; o.x = sx; o.y = sy; o.z = sz; o.w = sw;
  *(float4*)(mag + (size_t)slice * PP + p) = o;
}

// ---------------------------------------------------------------------------
// Kernel 2: per-slice selection + graph. One workgroup per slice.
// ---------------------------------------------------------------------------
#if __has_builtin(__builtin_amdgcn_tensor_load_to_lds)
#define TDM_FIRST_SLOT 8192   /* TDM stages float4-slots [0,8192) = rows 0..127 */
#else
#define TDM_FIRST_SLOT 0
#endif

__global__ void __launch_bounds__(256)
slice_kernel(const float* __restrict__ mag, float* __restrict__ graph) {
  __shared__ __align__(16) float magLds[PP];          // 256 KB staged slice
  __shared__ unsigned            hist[256];
  __shared__ unsigned            sScal[4];
  __shared__ unsigned long long  cand[256];
  __shared__ float               sCy[RTOP], sCx[RTOP];
  __shared__ float               sRed[5];
  __shared__ float               sIC[4];
  __shared__ __align__(16) float V0[4096];
  __shared__ __align__(16) float V1[4096];
  __shared__ float               M[4096];
  __shared__ unsigned            ties[256];

  const int tid   = threadIdx.x;
  const int lane  = tid & 31;
  const int wave  = tid >> 5;
  const int slice = blockIdx.x;
  const float* m  = mag + (size_t)slice * PP;
  const unsigned ldsBase = (unsigned)(size_t)&magLds[0];

  // ---- stage 256KB mag slice into LDS --------------------------------------
#if __has_builtin(__builtin_amdgcn_tensor_load_to_lds)
  if (wave == 0) {
    const unsigned long long ga = (unsigned long long)(size_t)m;
    u32x4  g0 = {0u, 0u, 0u, 0u};
    i32x8v g1 = {0, 0, 0, 0, 0, 0, 0, 0};
    i32x4v g2 = {0, 0, 0, 0};
    i32x4v g3 = {0, 0, 0, 0};
    // D# group0: count=1 | lds_addr | global_addr[56:0] | type=2
    g0[0] = 1u;
    g0[1] = ldsBase;
    g0[2] = (unsigned)(ga & 0xFFFFFFFFull);
    g0[3] = (unsigned)((ga >> 32) & 0x01FFFFFFull) | (2u << 30);
    // D# group1: data_size=4B; tensor 256x128; tile 256x128; stride0=256
    g1[0] = (int)(2u << 16);                 // data_size = 4 bytes
    g1[1] = (int)(256u << 16);               // tensor_dim0 = 256
    g1[2] = (int)(128u << 16);               // tensor_dim1 = 128
    g1[3] = (int)(256u << 16);               // tile_dim0  = 256
    g1[4] = (int)128;                        // tile_dim1  = 128 (tile_dim2=0)
    g1[5] = (int)256;                        // tensor_dim0_stride = 256
#if __clang_major__ >= 23
    i32x8v g4 = {0, 0, 0, 0, 0, 0, 0, 0};
    __builtin_amdgcn_tensor_load_to_lds(g0, g1, g2, g3, g4, 0);
#else
    __builtin_amdgcn_tensor_load_to_lds(g0, g1, g2, g3, 0);
#endif
  }
  __builtin_amdgcn_s_wait_tensorcnt(0);
#endif
  // remaining float4 slots via per-lane async copies (ASYNCcnt-tracked)
  for (int s = TDM_FIRST_SLOT + tid; s < PP / 4; s += 256) {
    const unsigned ldsoff = ldsBase + ((unsigned)s << 4);
    const unsigned long long gaddr = (unsigned long long)(size_t)(m + ((size_t)s << 2));
    asm volatile("global_load_async_to_lds_b128 %0, %1, off"
                 :: "v"(ldsoff), "v"(gaddr) : "memory");
  }
  asm volatile("s_wait_asynccnt 0" ::: "memory");
  __syncthreads();

  // ---- radix-select the exact 64th-largest key over 65536 values ----------
  if (tid == 0) { sScal[0] = 0u; sScal[1] = RTOP; }
  __syncthreads();
  for (int pass = 3; pass >= 0; --pass) {
    hist[tid] = 0u;
    __syncthreads();
    const unsigned prefix = sScal[0];
    const unsigned shift  = (unsigned)pass * 8u;
    const unsigned mask   = (pass == 3) ? 0u : (0xFFFFFFFFu << ((pass + 1) * 8));
    for (int i = tid; i < PP; i += 256) {
      const unsigned k = keyAsc(magLds[i]);
      if ((k & mask) == prefix) atomicAdd(&hist[(k >> shift) & 255u], 1u);
    }
    __syncthreads();
    if (tid == 0) {
      const unsigned want = sScal[1];
      unsigned cum = 0; int v = 255;
      for (; v >= 0; --v) { cum += hist[v]; if (cum >= want) break; }
      if (v < 0) v = 0;
      sScal[1] = want - (cum - hist[v]);
      sScal[0] = prefix | ((unsigned)v << shift);
    }
    __syncthreads();
  }
  const unsigned kth = sScal[0];

  // ---- collect candidates (key >= kth) -------------------------------------
  if (tid == 0) sScal[2] = 0u;
  cand[tid] = 0ull;
  __syncthreads();
  for (int i = tid; i < PP; i += 256) {
    const unsigned k = keyAsc(magLds[i]);
    if (k >= kth) {
      const unsigned pos = atomicAdd(&sScal[2], 1u);
      if (pos < 256u)
        cand[pos] = ((unsigned long long)k << 32) | (unsigned)(~(unsigned)i);
    }
  }
  __syncthreads();
  // bitonic sort, descending (equal keys: lower index first via ~idx)
  for (int kk = 2; kk <= 256; kk <<= 1) {
    for (int j = kk >> 1; j > 0; j >>= 1) {
      const int ixj = tid ^ j;
      if (ixj > tid) {
        const unsigned long long A = cand[tid], Bq = cand[ixj];
        const bool desc = ((tid & kk) == 0);
        if (desc ? (A < Bq) : (A > Bq)) { cand[tid] = Bq; cand[ixj] = A; }
      }
      __syncthreads();
    }
  }

  // ---- centers, mean, covariance, 2x2 inverse ------------------------------
  if (tid < 5) sRed[tid] = 0.f;
  if (tid < RTOP) {
    const unsigned idx = (~(unsigned)(cand[tid] & 0xFFFFFFFFull)) & 0xFFFFu;
    sCy[tid] = (float)((idx >> 6) * 8 + 4);
    sCx[tid] = (float)((idx & 63u) * 8 + 4);
  }
  __syncthreads();
  if (tid < RTOP) { atomicAdd(&sRed[0], sCy[tid]); atomicAdd(&sRed[1], sCx[tid]); }
  __syncthreads();
  const float mY = sRed[0] * (1.f / 64.f);
  const float mX = sRed[1] * (1.f / 64.f);
  if (tid < RTOP) {
    const float dy = sCy[tid] - mY, dx = sCx[tid] - mX;
    atomicAdd(&sRed[2], dy * dy);
    atomicAdd(&sRed[3], dy * dx);
    atomicAdd(&sRed[4], dx * dx);
  }
  __syncthreads();
  if (tid == 0) {
    const float a  = sRed[2] * (1.f / 63.f) + 1e-6f;
    const float bq = sRed[3] * (1.f / 63.f);
    const float c  = sRed[4] * (1.f / 63.f) + 1e-6f;
    const float inv = 1.f / (a * c - bq * bq);
    sIC[0] =  c * inv;  sIC[1] = -bq * inv;
    sIC[2] = -bq * inv; sIC[3] =  a * inv;
  }
  __syncthreads();

  // ---- v = diff @ inv_cov via V_WMMA_F32_16X16X4_F32 -----------------------
  const float ic00 = sIC[0], ic01 = sIC[1], ic10 = sIC[2], ic11 = sIC[3];
  for (int t = wave; t < 256; t += 8) {
    const int m0 = t << 4;
    const int q  = m0 >> 6;
    float e0 = 0.f, e1 = 0.f;
    if (lane < 16) {
      const int r = (m0 & 63) + lane;
      e0 = sCy[r] - sCy[q];
      e1 = sCx[r] - sCx[q];
    }
    float b0 = 0.f, b1 = 0.f;
    if (lane == 0) { b0 = ic00; b1 = ic10; }
    if (lane == 1) { b0 = ic01; b1 = ic11; }
    v2f Av = {e0, e1};
    v2f Bv = {b0, b1};
    v8f Dv = {};
#if __has_builtin(__builtin_amdgcn_wmma_f32_16x16x4_f32)
    Dv = __builtin_amdgcn_wmma_f32_16x16x4_f32(
        false, Av, false, Bv, (short)0, Dv, false, false);
    // D columns 0/1 hold v0/v1: lane {0,16} -> V0 rows, lane {1,17} -> V1 rows
    const int col = lane & 15;
    if (col < 2) {
      float* dst = (col == 0) ? V0 : V1;
      const int rb = m0 + ((lane >> 4) << 3);
      float4 lo; lo.x = Dv[0]; lo.y = Dv[1]; lo.z = Dv[2]; lo.w = Dv[3];
      float4 hi; hi.x = Dv[4]; hi.y = Dv[5]; hi.z = Dv[6]; hi.w = Dv[7];
      *(float4*)(dst + rb)     = lo;
      *(float4*)(dst + rb + 4) = hi;
    }
#else
    if (lane < 16) {
      V0[m0 + lane] = e0 * ic00 + e1 * ic10;
      V1[m0 + lane] = e0 * ic01 + e1 * ic11;
    }
#endif
  }
  __syncthreads();

  // maha = sqrt(v0*e0) + sqrt(v1*e1)
  for (int i = tid; i < 4096; i += 256) {
    const int q = i >> 6, r = i & 63;
    const float e0 = sCy[r] - sCy[q], e1 = sCx[r] - sCx[q];
    M[i] = sqrtf(V0[i] * e0) + sqrtf(V1[i] * e1);
  }
  __syncthreads();

  // ---- radix-select 169th-smallest maha ------------------------------------
  if (tid == 0) { sScal[0] = 0u; sScal[1] = TAUSQ; }
  __syncthreads();
  for (int pass = 3; pass >= 0; --pass) {
    hist[tid] = 0u;
    __syncthreads();
    const unsigned prefix = sScal[0];
    const unsigned shift  = (unsigned)pass * 8u;
    const unsigned mask   = (pass == 3) ? 0u : (0xFFFFFFFFu << ((pass + 1) * 8));
    for (int i = tid; i < 4096; i += 256) {
      const unsigned k = keyAsc(M[i]);
      if ((k & mask) == prefix) atomicAdd(&hist[(k >> shift) & 255u], 1u);
    }
    __syncthreads();
    if (tid == 0) {
      const unsigned want = sScal[1];
      unsigned cum = 0; int v = 0;
      for (; v < 256; ++v) { cum += hist[v]; if (cum >= want) break; }
      if (v > 255) v = 255;
      sScal[1] = want - (cum - hist[v]);
      sScal[0] = prefix | ((unsigned)v << shift);
    }
    __syncthreads();
  }
  const unsigned kth2 = sScal[0];

  if (tid == 0) { sScal[2] = 0u; sScal[3] = 0u; }
  __syncthreads();
  for (int i = tid; i < 4096; i += 256) {
    const unsigned k = keyAsc(M[i]);
    if (k < kth2) atomicAdd(&sScal[2], 1u);
    else if (k == kth2) {
      const unsigned p = atomicAdd(&sScal[3], 1u);
      if (p < 256u) ties[p] = (unsigned)i;
    }
  }
  __syncthreads();
  const unsigned nless = sScal[2];
  const unsigned ntie  = sScal[3] < 256u ? sScal[3] : 256u;
  const unsigned tsel  = TAUSQ - nless;

  float* og = graph + (size_t)slice * 4096;
  for (int i = tid; i < 4096; i += 256) {
    const unsigned k = keyAsc(M[i]);
    float val = 0.f;
    if (k < kth2) val = 1.f;
    else if (k == kth2) {
      unsigned rank = 0;
      for (unsigned j = 0; j < ntie; ++j) rank += (ties[j] < (unsigned)i) ? 1u : 0u;
      if (rank < tsel) val = 1.f;
    }
    og[i] = val;
  }
}

extern "C" void kernel_launch(void* const* d_in, const int* in_sizes, int n_in,
                              void* d_out, int out_size, void* d_ws, size_t ws_size,
                              hipStream_t stream) {
  const float* x = (const float*)d_in[0];      // (4,8,16,256,256) f32
  float* mag   = (float*)d_ws;                 // 28*65536 f32 = 7.34 MB scratch
  float* graph = (float*)d_out;                // (4,7,64,64) f32

  mag_kernel  <<<NSLICE * (PP / 4) / 256, 256, 0, stream>>>(x, mag);
  slice_kernel<<<NSLICE,                 256, 0, stream>>>(mag, graph);
}